// Loss_fn_66563403153931
// MI455X (gfx1250) — compile-verified
//
#include <hip/hip_runtime.h>
#include <hip/hip_bf16.h>

// ---------------------------------------------------------------------------
// EMD (auction) + repulsion loss for (8, 2048, 3) point clouds on MI455X.
//
// Design (MI455X-specific):
//  * benefit' matrix  2*p.g - |g|^2  (8 x 2048 x 2048 f32 = 134 MB) is built
//    once with V_WMMA_F32_16X16X4_F32 (K padded to 4, bias folded into K=3).
//    It fits in the 192 MB L2, so the 40 auction sweeps stream from L2.
//    The omitted -|p|^2 is a per-row constant: invariant for top-2 deltas,
//    bids, per-column winner argmax and the fallback row argmax => matching
//    is exact.
//  * Auction: one 1024-thread workgroup per batch; price/owner/assign and
//    packed (bid,i) column-max all in LDS; 64-bit LDS atomicMax implements
//    the per-column winner selection (positive floats order as uints).
//  * Repulsion: per-batch points in LDS (float4-padded), register 10-NN
//    insertion per point.
// Workspace: needs 8*2048*2048*4 = 134,217,728 bytes in d_ws.
// ---------------------------------------------------------------------------

#define BN 8
#define NP 2048
#define NT (NP / 16)          // 128 tiles of 16 along each point axis
#define AUCTION_ITERS 40
#define AEPS 0.01f
#define RL_H 0.0005f
#define NEG_BIG (-1.0e30f)

typedef float v2f __attribute__((ext_vector_type(2)));
typedef float v8f __attribute__((ext_vector_type(8)));

__device__ __forceinline__ float block_reduce_sum(float v, float* redbuf) {
  #pragma unroll
  for (int off = 16; off > 0; off >>= 1) v += __shfl_down(v, off, 32);
  const int lane = threadIdx.x & 31;
  const int wave = threadIdx.x >> 5;
  if (lane == 0) redbuf[wave] = v;
  __syncthreads();
  const int nw = blockDim.x >> 5;
  v = (threadIdx.x < (unsigned)nw) ? redbuf[threadIdx.x] : 0.0f;
  if (wave == 0) {
    #pragma unroll
    for (int off = 16; off > 0; off >>= 1) v += __shfl_down(v, off, 32);
  }
  return v;  // valid in thread 0
}

// ---------------------------------------------------------------------------
// Kernel 1: benefit'[b,i,j] = 2*pred_i . gt_j - |gt_j|^2 via WMMA f32 16x16x4.
//   A (16x4) row i-tile: [2px, 2py, 2pz, 1]
//   B (4x16) col j-tile: [gx, gy, gz, -|g|^2]^T
// Fragment lane layout (ISA 7.12.2, 32-bit 16x4): VGPR0 = K0 | K2,
// VGPR1 = K1 | K3 (low half-wave | high half-wave), same split for B.
// ---------------------------------------------------------------------------
__global__ __launch_bounds__(256) void benefit_wmma_kernel(
    const float* __restrict__ pred, const float* __restrict__ gt,
    float* __restrict__ benefit, float* __restrict__ out) {
  if (blockIdx.x == 0 && threadIdx.x == 0) { out[0] = 0.0f; out[1] = 0.0f; }

  const int b     = blockIdx.x / NT;
  const int itile = blockIdx.x % NT;
  const int lane  = threadIdx.x & 31;
  const int wave  = threadIdx.x >> 5;          // 8 waves per block
  const int m     = lane & 15;                 // A row / B col within tile
  const bool hi   = (lane >= 16);

  // A fragment: lanes 0-15 hold (K0,K1)=(2px,2py); lanes 16-31 hold (K2,K3)=(2pz,1)
  const float* pb = pred + ((size_t)b * NP + (size_t)itile * 16 + m) * 3;
  const float px = pb[0], py = pb[1], pz = pb[2];
  v2f afrag;
  afrag.x = hi ? (2.0f * pz) : (2.0f * px);
  afrag.y = hi ? 1.0f        : (2.0f * py);

  for (int jt = wave; jt < NT; jt += 8) {
    const float* gb = gt + ((size_t)b * NP + (size_t)jt * 16 + m) * 3;
    const float gx = gb[0], gy = gb[1], gz = gb[2];
    const float gn = gx * gx + gy * gy + gz * gz;
    v2f bfrag;
    bfrag.x = hi ? gz    : gx;
    bfrag.y = hi ? (-gn) : gy;

    v8f acc = {0.0f, 0.0f, 0.0f, 0.0f, 0.0f, 0.0f, 0.0f, 0.0f};
    acc = __builtin_amdgcn_wmma_f32_16x16x4_f32(
        /*neg_a=*/false, afrag, /*neg_b=*/false, bfrag,
        /*c_mod=*/(short)0, acc, /*reuse_a=*/false, /*reuse_b=*/false);

    // D layout: VGPR r -> row r (lanes 0-15) / row r+8 (lanes 16-31), col = m
    float* drow = benefit +
        ((size_t)b * NP + (size_t)itile * 16 + (hi ? 8 : 0)) * (size_t)NP +
        (size_t)jt * 16 + m;
    #pragma unroll
    for (int r = 0; r < 8; ++r) drow[(size_t)r * NP] = acc[r];
  }
}

// ---------------------------------------------------------------------------
// Kernel 2: Jacobi auction, one workgroup per batch, state in LDS.
// ---------------------------------------------------------------------------
__global__ __launch_bounds__(1024) void auction_kernel(
    const float* __restrict__ benefit, const float* __restrict__ pred,
    const float* __restrict__ gt, float* __restrict__ out) {
  __shared__ float price[NP];
  __shared__ int owner[NP];
  __shared__ int assign[NP];
  __shared__ unsigned long long packed[NP];
  __shared__ float redbuf[32];

  const int b = blockIdx.x;
  const int tid = threadIdx.x;
  const float* bmat = benefit + (size_t)b * NP * NP;

  for (int j = tid; j < NP; j += 1024) { price[j] = 0.0f; owner[j] = -1; }
  __syncthreads();

  for (int iter = 0; iter < AUCTION_ITERS; ++iter) {
    for (int i = tid; i < NP; i += 1024) assign[i] = -1;
    __syncthreads();
    for (int j = tid; j < NP; j += 1024) {
      packed[j] = 0ull;
      const int o = owner[j];
      if (o >= 0) assign[o] = j;
    }
    __syncthreads();

    // Bidding: each unassigned row scans its (L2-resident) benefit row.
    for (int i = tid; i < NP; i += 1024) {
      if (assign[i] >= 0) continue;
      const float4* row = (const float4*)(bmat + (size_t)i * NP);
      float v1 = NEG_BIG, v2 = NEG_BIG;
      int j1 = 0;
      for (int j4 = 0; j4 < NP / 4; ++j4) {
        const float4 r = row[j4];
        const int j = j4 * 4;
        float v;
        v = r.x - price[j + 0]; if (v > v1) { v2 = v1; v1 = v; j1 = j + 0; } else if (v > v2) v2 = v;
        v = r.y - price[j + 1]; if (v > v1) { v2 = v1; v1 = v; j1 = j + 1; } else if (v > v2) v2 = v;
        v = r.z - price[j + 2]; if (v > v1) { v2 = v1; v1 = v; j1 = j + 2; } else if (v > v2) v2 = v;
        v = r.w - price[j + 3]; if (v > v1) { v2 = v1; v1 = v; j1 = j + 3; } else if (v > v2) v2 = v;
      }
      const float bid = price[j1] + (v1 - v2) + AEPS;  // > 0 always
      const unsigned long long pk =
          ((unsigned long long)__float_as_uint(bid) << 32) |
          (unsigned long long)(unsigned)(NP - 1 - i);   // smallest-i tiebreak
      atomicMax(&packed[j1], pk);
    }
    __syncthreads();

    for (int j = tid; j < NP; j += 1024) {
      const unsigned long long pk = packed[j];
      if (pk != 0ull) {
        price[j] = __uint_as_float((unsigned)(pk >> 32));
        owner[j] = NP - 1 - (int)(unsigned)(pk & 0xffffffffull);
      }
    }
    __syncthreads();
  }

  // Final assignment (+ fallback argmax) and matched squared-distance sum.
  for (int i = tid; i < NP; i += 1024) assign[i] = -1;
  __syncthreads();
  for (int j = tid; j < NP; j += 1024) {
    const int o = owner[j];
    if (o >= 0) assign[o] = j;
  }
  __syncthreads();

  float local = 0.0f;
  for (int i = tid; i < NP; i += 1024) {
    int jm = assign[i];
    if (jm < 0) {  // fallback: argmax_j (benefit - price); row-const invariant
      const float4* row = (const float4*)(bmat + (size_t)i * NP);
      float v1 = NEG_BIG;
      jm = 0;
      for (int j4 = 0; j4 < NP / 4; ++j4) {
        const float4 r = row[j4];
        const int j = j4 * 4;
        float v;
        v = r.x - price[j + 0]; if (v > v1) { v1 = v; jm = j + 0; }
        v = r.y - price[j + 1]; if (v > v1) { v1 = v; jm = j + 1; }
        v = r.z - price[j + 2]; if (v > v1) { v1 = v; jm = j + 2; }
        v = r.w - price[j + 3]; if (v > v1) { v1 = v; jm = j + 3; }
      }
    }
    const float* pp = pred + ((size_t)b * NP + i) * 3;
    const float* gg = gt + ((size_t)b * NP + jm) * 3;
    const float dx = pp[0] - gg[0];
    const float dy = pp[1] - gg[1];
    const float dz = pp[2] - gg[2];
    local += dx * dx + dy * dy + dz * dz;
  }
  const float total = block_reduce_sum(local, redbuf);
  if (tid == 0)
    atomicAdd(out + 0, total * (1.0f / ((float)BN * (float)NP * 3.0f)));
}

// ---------------------------------------------------------------------------
// Kernel 3: repulsion loss, 10-NN per point, points in LDS.
// ---------------------------------------------------------------------------
__global__ __launch_bounds__(256) void repulsion_kernel(
    const float* __restrict__ x, float* __restrict__ out) {
  __shared__ float4 pts[NP];   // 32 KB, float4-padded for bank friendliness
  __shared__ float redbuf[32];

  const int b = blockIdx.x >> 3;      // 8 blocks per batch
  const int seg = blockIdx.x & 7;     // 256 points per block
  for (int j = threadIdx.x; j < NP; j += 256) {
    const float* p = x + ((size_t)b * NP + j) * 3;
    pts[j] = make_float4(p[0], p[1], p[2], 0.0f);
  }
  __syncthreads();

  const int i = seg * 256 + threadIdx.x;
  const float4 pi = pts[i];

  float nn[10];
  #pragma unroll
  for (int m = 0; m < 10; ++m) nn[m] = 3.4e38f;

  for (int j = 0; j < NP; ++j) {       // lockstep j => LDS broadcast reads
    const float4 pj = pts[j];
    const float dx = pi.x - pj.x;
    const float dy = pi.y - pj.y;
    const float dz = pi.z - pj.z;
    const float d2 = dx * dx + dy * dy + dz * dz;   // j==i contributes 0 (self)
    if (d2 < nn[9]) {
      nn[9] = d2;
      #pragma unroll
      for (int m = 9; m > 0; --m) {
        const float lo = fminf(nn[m - 1], nn[m]);
        const float hv = fmaxf(nn[m - 1], nn[m]);
        nn[m - 1] = lo;
        nn[m] = hv;
      }
    }
  }

  float s = 0.0f;
  #pragma unroll
  for (int m = 1; m < 10; ++m) s += fmaxf(RL_H - nn[m], 0.0f);  // skip self

  const float total = block_reduce_sum(s, redbuf);
  if (threadIdx.x == 0)
    atomicAdd(out + 1, total * (1.0f / ((float)BN * (float)NP * 9.0f)));
}

// ---------------------------------------------------------------------------
extern "C" void kernel_launch(void* const* d_in, const int* in_sizes, int n_in,
                              void* d_out, int out_size, void* d_ws, size_t ws_size,
                              hipStream_t stream) {
  const float* x = (const float*)d_in[0];    // pred (8,2048,3)
  const float* gt = (const float*)d_in[1];   // gt   (8,2048,3)
  float* out = (float*)d_out;                // [loss_cd, loss_rl]
  float* benefit = (float*)d_ws;             // 8*2048*2048 f32 = 134 MB

  benefit_wmma_kernel<<<BN * NT, 256, 0, stream>>>(x, gt, benefit, out);
  auction_kernel<<<BN, 1024, 0, stream>>>(benefit, x, gt, out);
  repulsion_kernel<<<BN * 8, 256, 0, stream>>>(x, out);
}